// PhysicsGuidedAttention_29403346108944
// MI455X (gfx1250) — compile-verified
//
#include <hip/hip_runtime.h>
#include <math.h>

// ---------------------------------------------------------------------------
// PhysicsGuidedAttention for MI455X (gfx1250, wave32, WMMA + async-to-LDS)
//   B=2, N=2048, D=768, H=12, Hd=64
//   fp32 math via V_WMMA_F32_16X16X4_F32 (bit-compatible with fp32 reference)
//   Operand streams (W, K, V) staged through LDS with double-buffered
//   GLOBAL_LOAD_ASYNC_TO_LDS_B128 (ASYNCcnt), 8 waves/block sharing each tile.
// ---------------------------------------------------------------------------

#define BATCH     2
#define SEQ       2048
#define DIM       768
#define NHEADS    12
#define HDIM      64
#define QKV_COLS  (3 * DIM)          // 2304
#define MROWS     (BATCH * SEQ)      // 4096
#define SCALE_F   0.125f             // 64^-0.5
#define INV_ELEV  0.001f             // 1/1000

#define WSTRIDE   36                 // padded k-stride (dwords) for W panels
#define KVSTRIDE  68                 // padded row stride (dwords) for K/V tiles

typedef float v2f __attribute__((ext_vector_type(2)));
typedef float v4f __attribute__((ext_vector_type(4)));
typedef float v8f __attribute__((ext_vector_type(8)));
typedef int   pga_v4i __attribute__((vector_size(16)));

#if defined(__HIP_DEVICE_COMPILE__) && \
    __has_builtin(__builtin_amdgcn_global_load_async_to_lds_b128) && \
    __has_builtin(__builtin_amdgcn_s_wait_asynccnt)
#define PGA_ASYNC 1
#else
#define PGA_ASYNC 0
#endif

#define PGA_AS1 __attribute__((address_space(1)))
#define PGA_AS3 __attribute__((address_space(3)))

// Move 16 bytes global -> LDS. Async (ASYNCcnt) when available.
static __device__ __forceinline__ void stage16(const float* g, float* l) {
#if PGA_ASYNC
    __builtin_amdgcn_global_load_async_to_lds_b128(
        (PGA_AS1 pga_v4i*)g, (PGA_AS3 pga_v4i*)l, 0, 0);
#else
    *(v4f*)l = *(const v4f*)g;
#endif
}
static __device__ __forceinline__ void stage_wait_tail2() {
#if PGA_ASYNC
    __builtin_amdgcn_s_wait_asynccnt(2);   // oldest in-flight tile complete
#endif
}
static __device__ __forceinline__ void stage_wait_all() {
#if PGA_ASYNC
    __builtin_amdgcn_s_wait_asynccnt(0);
#endif
}

// D(16x16,f32) = A(16x4,f32) x B(4x16,f32) + C
static __device__ __forceinline__ v8f wmma4(v2f a, v2f b, v8f c) {
    return __builtin_amdgcn_wmma_f32_16x16x4_f32(
        false, a, false, b, (short)0, c, false, false);
}

// ---------------------------------------------------------------------------
// Kernel 1: qkv = x @ w_qkv^T ; scatter into q/k/v as [B, H, N, Hd].
// Block = 8 waves = 128 rows x 64 cols; W k-panels (64 cols x 32 k) streamed
// through double-buffered LDS, shared by all 8 waves.
// A-layout: lane = (m&15), comps k = kbase + 2*(lane>>4) + {0,1}; B mirrors A.
// ---------------------------------------------------------------------------
__global__ void __launch_bounds__(256) pga_qkv_gemm(
        const float* __restrict__ x, const float* __restrict__ w_qkv,
        float* __restrict__ q, float* __restrict__ k, float* __restrict__ v)
{
    __shared__ float wbuf[2][64 * WSTRIDE];

    const int tid  = threadIdx.x;
    const int lane = tid & 31;
    const int wv   = tid >> 5;
    const int tilesN = QKV_COLS / 64;            // 36
    const int bn = blockIdx.x % tilesN;
    const int bm = blockIdx.x / tilesN;          // 0..31
    const int n0 = bn * 64;
    const int m0 = bm * 128 + wv * 16;
    const int nl = lane & 15;
    const int g  = lane >> 4;
    const int kg = g * 2;

    // Staging map: 512 float4 per panel, 2 per thread.
    const int f0 = tid, f1 = tid + 256;
    const int c0 = f0 >> 3, k40 = (f0 & 7) << 2;
    const int c1 = f1 >> 3, k41 = (f1 & 7) << 2;
    const float* __restrict__ gw0 = w_qkv + (size_t)(n0 + c0) * DIM + k40;
    const float* __restrict__ gw1 = w_qkv + (size_t)(n0 + c1) * DIM + k41;
    float* lw0 = &wbuf[0][c0 * WSTRIDE + k40];
    float* lw1 = &wbuf[0][c1 * WSTRIDE + k41];
    const int LBUF = 64 * WSTRIDE;

    const int NP = DIM / 32;                     // 24 k-panels
    // Prologue: stage panels 0,1
    stage16(gw0, lw0);             stage16(gw1, lw1);
    stage16(gw0 + 32, lw0 + LBUF); stage16(gw1 + 32, lw1 + LBUF);

    v8f acc[4] = {};
    const float* __restrict__ arow = x + (size_t)(m0 + nl) * DIM + kg;

    for (int p = 0; p < NP; ++p) {
        const int cur = p & 1;
        if (p + 2 < NP) stage_wait_tail2(); else stage_wait_all();
        __syncthreads();
        const float* __restrict__ wp = &wbuf[cur][0];
        const int pb = p * 32;
#pragma unroll
        for (int kk = 0; kk < 32; kk += 4) {
            v2f a = *(const v2f*)(arow + pb + kk);
#pragma unroll
            for (int j = 0; j < 4; ++j) {
                v2f b = *(const v2f*)(wp + (j * 16 + nl) * WSTRIDE + kk + kg);
                acc[j] = wmma4(a, b, acc[j]);
            }
        }
        __syncthreads();
        if (p + 2 < NP) {
            const int off = (p + 2) * 32;
            stage16(gw0 + off, lw0 + cur * LBUF);
            stage16(gw1 + off, lw1 + cur * LBUF);
        }
    }

    // Scatter: whole 64-wide tile lands in one (which, head).
    const int which = n0 / DIM;
    const int h     = (n0 % DIM) / HDIM;
    float* __restrict__ dst = (which == 0) ? q : (which == 1) ? k : v;
    const int b   = m0 / SEQ;
    const int nr0 = m0 % SEQ;
    const size_t headbase = ((size_t)b * NHEADS + h) * SEQ;

#pragma unroll
    for (int j = 0; j < 4; ++j) {
        const int hd = j * 16 + nl;
#pragma unroll
        for (int r = 0; r < 8; ++r) {
            const int row = nr0 + r + 8 * g;
            dst[(headbase + row) * HDIM + hd] = acc[j][r];
        }
    }
}

// ---------------------------------------------------------------------------
// Kernel 2: flash attention. Block = one (b,h) x 8 consecutive query tiles;
// the 16x64 K and V chunks are streamed once per block through double-buffered
// LDS (8x less L2 traffic than per-wave streaming). S^T = K@Q^T so softmax
// reductions over keys are register reductions; online softmax; O += P@V via
// shfl_xor(16)-based P^T -> A-fragment relayout.
// ---------------------------------------------------------------------------
__global__ void __launch_bounds__(256) pga_flash_attn(
        const float* __restrict__ q, const float* __restrict__ k,
        const float* __restrict__ v, const float* __restrict__ elev,
        const float* __restrict__ alpha_p, float* __restrict__ aout)
{
    __shared__ float kbuf[2][16 * KVSTRIDE];
    __shared__ float vbuf[2][16 * KVSTRIDE];

    const int tid  = threadIdx.x;
    const int lane = tid & 31;
    const int wv   = tid >> 5;
    const int qb = blockIdx.x % 16;                       // 8-tile group
    const int hh = (blockIdx.x / 16) % NHEADS;
    const int b  = blockIdx.x / (16 * NHEADS);
    const int q0 = (qb * 8 + wv) * 16;
    const int nl = lane & 15;
    const int g  = lane >> 4;

    const size_t bh = ((size_t)b * NHEADS + hh) * SEQ;
    const float* __restrict__ Kb = k + bh * HDIM;
    const float* __restrict__ Vb = v + bh * HDIM;
    const float* __restrict__ eb = elev + (size_t)b * SEQ;

    // Staging map: 256 float4 per matrix, 1 per thread each for K and V.
    const int srow = tid >> 4;
    const int sc4  = (tid & 15) << 2;
    const float* __restrict__ gK = Kb + (size_t)srow * HDIM + sc4;
    const float* __restrict__ gV = Vb + (size_t)srow * HDIM + sc4;
    float* lK = &kbuf[0][srow * KVSTRIDE + sc4];
    float* lV = &vbuf[0][srow * KVSTRIDE + sc4];
    const int LBUF = 16 * KVSTRIDE;
    const int CH_G = 16 * HDIM;                           // chunk stride (floats)

    // Preload Q tile as B-fragments: lane holds row (q0+nl), comps d = 4t+2g.
    v2f qf[16];
    {
        const float* __restrict__ Qr = q + (bh + q0 + nl) * HDIM + 2 * g;
#pragma unroll
        for (int t = 0; t < 16; ++t) qf[t] = *(const v2f*)(Qr + 4 * t);
    }
    const float alpha = alpha_p[0];
    const float ei    = eb[q0 + nl];

    const int NC = SEQ / 16;                              // 128 chunks
    // Prologue: stage chunks 0,1
    stage16(gK, lK);               stage16(gV, lV);
    stage16(gK + CH_G, lK + LBUF); stage16(gV + CH_G, lV + LBUF);

    float runmax = -INFINITY, runsum = 0.0f;
    v8f o[4] = {};

    for (int ci = 0; ci < NC; ++ci) {
        const int cur = ci & 1;
        const int j0  = ci * 16;
        if (ci + 2 < NC) stage_wait_tail2(); else stage_wait_all();
        __syncthreads();
        const float* __restrict__ kc = &kbuf[cur][0];
        const float* __restrict__ vc = &vbuf[cur][0];

        // ---- S^T = K_chunk @ Q^T (A row = key = lane&15, comps along d) ----
        v8f st = {};
        const float* __restrict__ Kr = kc + nl * KVSTRIDE + 2 * g;
#pragma unroll
        for (int t = 0; t < 16; ++t) {
            v2f a = *(const v2f*)(Kr + 4 * t);
            st = wmma4(a, qf[t], st);
        }
        // st[r] = S^T[key = j0 + r + 8g][query m = nl]
        float p[8];
        float cmax = -INFINITY;
#pragma unroll
        for (int r = 0; r < 8; ++r) {
            const float ej   = eb[j0 + r + 8 * g];
            const float diff = (ej - ei) * INV_ELEV;
            const float bias = -fminf(alpha * fmaxf(diff, 0.0f), 10.0f);
            const float s    = st[r] * SCALE_F + bias;
            p[r] = s;
            cmax = fmaxf(cmax, s);
        }
        cmax = fmaxf(cmax, __shfl_xor(cmax, 16, 32));
        const float newmax = fmaxf(runmax, cmax);
        const float corr   = __expf(runmax - newmax);
        runmax = newmax;

        float csum = 0.0f;
#pragma unroll
        for (int r = 0; r < 8; ++r) { p[r] = __expf(p[r] - newmax); csum += p[r]; }
        csum  += __shfl_xor(csum, 16, 32);
        runsum = runsum * corr + csum;

        // Rescale O rows (row m = r + 8g) by corr[m] (corr lives at lane m).
#pragma unroll
        for (int r = 0; r < 8; ++r) {
            const float fr = __shfl(corr, r + 8 * g, 32);
            o[0][r] *= fr; o[1][r] *= fr; o[2][r] *= fr; o[3][r] *= fr;
        }

        // ---- O += P @ V.  A = P(16x16): P^T[kk][nl] lives in reg kk%8 at
        //      lane-group kk/8. kk = 4s + 2g + {0,1}; srcgroup = (s>=2). ----
#pragma unroll
        for (int s = 0; s < 4; ++s) {
            const float A0 = p[(4 * s)     & 7], A1 = p[(4 * s + 1) & 7];
            const float B0 = p[(4 * s + 2) & 7], B1 = p[(4 * s + 3) & 7];
            v2f af;
            if (s < 2) {  // source lane-group 0
                const float B0x = __shfl_xor(B0, 16, 32);
                const float B1x = __shfl_xor(B1, 16, 32);
                af.x = (g == 0) ? A0 : B0x;
                af.y = (g == 0) ? A1 : B1x;
            } else {      // source lane-group 1
                const float A0x = __shfl_xor(A0, 16, 32);
                const float A1x = __shfl_xor(A1, 16, 32);
                af.x = (g == 0) ? A0x : B0;
                af.y = (g == 0) ? A1x : B1;
            }
            const int kb = 4 * s + 2 * g;
            const float* __restrict__ Vr = vc + kb * KVSTRIDE;
#pragma unroll
            for (int dt = 0; dt < 4; ++dt) {
                v2f bf;
                bf.x = Vr[dt * 16 + nl];
                bf.y = Vr[KVSTRIDE + dt * 16 + nl];
                o[dt] = wmma4(af, bf, o[dt]);
            }
        }

        __syncthreads();
        if (ci + 2 < NC) {
            const size_t off = (size_t)(ci + 2) * CH_G;
            stage16(gK + off, lK + cur * LBUF);
            stage16(gV + off, lV + cur * LBUF);
        }
    }

    // Epilogue: divide by row sums, write [B, N, H*Hd] (contiguous for proj).
    const float rinv = 1.0f / runsum;
#pragma unroll
    for (int r = 0; r < 8; ++r) {
        const float fr = __shfl(rinv, r + 8 * g, 32);
        const int   m  = q0 + r + 8 * g;
        const size_t rowoff = ((size_t)b * SEQ + m) * DIM + hh * HDIM;
#pragma unroll
        for (int dt = 0; dt < 4; ++dt)
            aout[rowoff + dt * 16 + nl] = o[dt][r] * fr;
    }
}

// ---------------------------------------------------------------------------
// Kernel 3: out = attn_out @ w_proj^T + b_proj. Same LDS-staged GEMM.
// ---------------------------------------------------------------------------
__global__ void __launch_bounds__(256) pga_proj_gemm(
        const float* __restrict__ a, const float* __restrict__ w_proj,
        const float* __restrict__ b_proj, float* __restrict__ out)
{
    __shared__ float wbuf[2][64 * WSTRIDE];

    const int tid  = threadIdx.x;
    const int lane = tid & 31;
    const int wv   = tid >> 5;
    const int tilesN = DIM / 64;                 // 12
    const int bn = blockIdx.x % tilesN;
    const int bm = blockIdx.x / tilesN;          // 0..31
    const int n0 = bn * 64;
    const int m0 = bm * 128 + wv * 16;
    const int nl = lane & 15;
    const int g  = lane >> 4;
    const int kg = g * 2;

    const int f0 = tid, f1 = tid + 256;
    const int c0 = f0 >> 3, k40 = (f0 & 7) << 2;
    const int c1 = f1 >> 3, k41 = (f1 & 7) << 2;
    const float* __restrict__ gw0 = w_proj + (size_t)(n0 + c0) * DIM + k40;
    const float* __restrict__ gw1 = w_proj + (size_t)(n0 + c1) * DIM + k41;
    float* lw0 = &wbuf[0][c0 * WSTRIDE + k40];
    float* lw1 = &wbuf[0][c1 * WSTRIDE + k41];
    const int LBUF = 64 * WSTRIDE;

    const int NP = DIM / 32;
    stage16(gw0, lw0);             stage16(gw1, lw1);
    stage16(gw0 + 32, lw0 + LBUF); stage16(gw1 + 32, lw1 + LBUF);

    v8f acc[4] = {};
    const float* __restrict__ arow = a + (size_t)(m0 + nl) * DIM + kg;

    for (int p = 0; p < NP; ++p) {
        const int cur = p & 1;
        if (p + 2 < NP) stage_wait_tail2(); else stage_wait_all();
        __syncthreads();
        const float* __restrict__ wp = &wbuf[cur][0];
        const int pb = p * 32;
#pragma unroll
        for (int kk = 0; kk < 32; kk += 4) {
            v2f av = *(const v2f*)(arow + pb + kk);
#pragma unroll
            for (int j = 0; j < 4; ++j) {
                v2f bv = *(const v2f*)(wp + (j * 16 + nl) * WSTRIDE + kk + kg);
                acc[j] = wmma4(av, bv, acc[j]);
            }
        }
        __syncthreads();
        if (p + 2 < NP) {
            const int off = (p + 2) * 32;
            stage16(gw0 + off, lw0 + cur * LBUF);
            stage16(gw1 + off, lw1 + cur * LBUF);
        }
    }

#pragma unroll
    for (int j = 0; j < 4; ++j) {
        const int gcol = n0 + j * 16 + nl;
        const float bb = b_proj[gcol];
#pragma unroll
        for (int r = 0; r < 8; ++r) {
            const int m = m0 + r + 8 * g;
            out[(size_t)m * DIM + gcol] = acc[j][r] + bb;
        }
    }
}

// ---------------------------------------------------------------------------
// Host launcher
// ---------------------------------------------------------------------------
extern "C" void kernel_launch(void* const* d_in, const int* in_sizes, int n_in,
                              void* d_out, int out_size, void* d_ws, size_t ws_size,
                              hipStream_t stream) {
    const float* x      = (const float*)d_in[0];
    const float* elev   = (const float*)d_in[1];
    const float* w_qkv  = (const float*)d_in[2];
    const float* w_proj = (const float*)d_in[3];
    const float* b_proj = (const float*)d_in[4];
    const float* alpha  = (const float*)d_in[5];
    float* out = (float*)d_out;

    const size_t per = (size_t)BATCH * NHEADS * SEQ * HDIM;   // 3,145,728 floats
    float* q    = (float*)d_ws;
    float* k    = q + per;
    float* v    = k + per;
    float* aout = v + per;
    (void)in_sizes; (void)n_in; (void)out_size; (void)ws_size;

    // Kernel 1: 32 x 36 blocks (each 128 rows x 64 cols)
    pga_qkv_gemm<<<dim3(32 * (QKV_COLS / 64)), dim3(256), 0, stream>>>(x, w_qkv, q, k, v);
    // Kernel 2: B * H * 16 blocks (each 8 query tiles of one head)
    pga_flash_attn<<<dim3(BATCH * NHEADS * 16), dim3(256), 0, stream>>>(q, k, v, elev, alpha, aout);
    // Kernel 3: 32 x 12 blocks
    pga_proj_gemm<<<dim3(32 * (DIM / 64)), dim3(256), 0, stream>>>(aout, w_proj, b_proj, out);
}